// PredictionAwareSAE_80393197846563
// MI455X (gfx1250) — compile-verified
//
#include <hip/hip_runtime.h>
#include <hip/hip_bf16.h>
#include <math.h>
#include <stdint.h>

// ---------------------------------------------------------------------------
// TopK Sparse Autoencoder forward for MI455X (gfx1250), wave32 + WMMA f32.
//   hb[h]    = latent_bias[h] - dot(pre_bias, W_enc[h,:])      (preamble)
//   pre_act  = x @ W_enc^T + hb                                [8192,16384]
//   latents  = scatter(relu(top32(pre_act)))                   [8192,16384]
//   recon    = latents @ W_enc + pre_bias                      [8192,768]
// GEMM tiles are DMA'd into LDS by the Tensor Data Mover (double-buffered),
// with D# padding producing the 34-float row stride the WMMA fragment
// addressing wants.  Falls back to manual copies if the TDM builtin is absent.
// ---------------------------------------------------------------------------

#define B_ROWS   8192
#define IN_DIM   768
#define HID_DIM  16384
#define TOPK     32

#define BM 128
#define BN 128
#define KC 32
#define LDSS 34                    // padded LDS row stride (floats)
#define TILE_F (BM * LDSS)         // floats per staged tile (4352)

typedef __attribute__((ext_vector_type(2))) float v2f;
typedef __attribute__((ext_vector_type(8))) float v8f;
typedef __attribute__((ext_vector_type(4))) unsigned int u32x4;
typedef __attribute__((ext_vector_type(8))) int i32x8;
typedef __attribute__((ext_vector_type(4))) int i32x4;

#if defined(__has_builtin)
#if __has_builtin(__builtin_amdgcn_tensor_load_to_lds) && \
    __has_builtin(__builtin_amdgcn_s_wait_tensorcnt)
#define USE_TDM 1
#endif
#endif

// ------------------------ Kernel 0: bias folding ---------------------------
// hb[h] = latent_bias[h] - dot(pre_bias, W[h,:]); one wave32 per h.

__global__ __launch_bounds__(256)
void sae_hbias(const float* __restrict__ W,
               const float* __restrict__ pre_bias,
               const float* __restrict__ latent_bias,
               float* __restrict__ hb)
{
    const int lane = threadIdx.x & 31;
    const int h    = blockIdx.x * 8 + (threadIdx.x >> 5);
    const float* wr = W + (size_t)h * IN_DIM;
    float s = 0.f;
    for (int d = lane; d < IN_DIM; d += 32) s += pre_bias[d] * wr[d];
#pragma unroll
    for (int off = 16; off > 0; off >>= 1) s += __shfl_down(s, off, 32);
    if (lane == 0) hb[h] = latent_bias[h] - s;
}

// ---------------------------- TDM tile loader ------------------------------
#if defined(USE_TDM)
// DMA a [128 rows x 32 cols] f32 tile (row stride IN_DIM in global) into LDS
// at byte offset lds_off, padded to a 34-float LDS row stride.
__device__ __forceinline__ void tdm_load_tile(unsigned lds_off,
                                              const float* gptr,
                                              unsigned tensor_rows)
{
    const unsigned long long ga = (unsigned long long)(uintptr_t)gptr;
    u32x4 g0;
    g0.x = 1u;                                          // count=1 (valid D#)
    g0.y = lds_off;                                     // lds_addr (bytes)
    g0.z = (unsigned)(ga & 0xffffffffu);                // global_addr[31:0]
    g0.w = (unsigned)((ga >> 32) & 0x01ffffffu)         // global_addr[56:32]
         | (2u << 30);                                  // type=2 ("image")
    i32x8 g1;
    g1[0] = (2 << 16)            // data_size = 4 bytes
          | (1 << 20)            // pad_enable
          | (4 << 22)            // pad_interval: every 32 DWORDs (one row)
          | (1 << 25);           // pad_amount: 2 DWORDs -> 34-float stride
    g1[1] = (int)((unsigned)IN_DIM << 16);              // tensor_dim0[15:0]
    g1[2] = (int)((tensor_rows & 0xffffu) << 16);       // dim0 hi=0 | dim1 lo
    g1[3] = (int)((tensor_rows >> 16) | ((unsigned)KC << 16)); // dim1 hi | tile_dim0
    g1[4] = BM;                                         // tile_dim1=128, tile_dim2=0
    g1[5] = IN_DIM;                                     // tensor_dim0_stride[31:0]
    g1[6] = 0;                                          // stride hi | dim1_stride lo
    g1[7] = 0;
    const i32x4 z4i = {0, 0, 0, 0};                     // groups 2/3 unused (2D)
    const i32x8 z8i = {0, 0, 0, 0, 0, 0, 0, 0};
    __builtin_amdgcn_tensor_load_to_lds(g0, g1, z4i, z4i, z8i, 0);
}
#endif

// ---------------------------- Kernel 1: encode GEMM ------------------------
// C-tile 128x128 per 256-thread block; 8 waves in 2(M) x 4(N) grid; each wave
// owns 4x2 WMMA 16x16 tiles.  K staged in LDS chunks of 32 (stride 34),
// double-buffered via TDM.

__global__ __launch_bounds__(256)
void sae_encode_wmma(const float* __restrict__ x,
                     const float* __restrict__ W,
                     const float* __restrict__ hb,
                     float* __restrict__ pre_act)
{
    extern __shared__ float lds[];   // [2 bufs][A tile | B tile], TILE_F each

    const int tid  = threadIdx.x;
    const int lane = tid & 31;
    const int wave = tid >> 5;
    const int wm   = wave >> 2;          // 0..1  (M direction)
    const int wn   = wave & 3;           // 0..3  (N direction)
    const int m0   = blockIdx.x * BM;
    const int n0   = blockIdx.y * BN;

    const int lrow = lane & 15;
    const int koff = (lane < 16) ? 0 : 2;

    v8f acc[4][2];
#pragma unroll
    for (int mi = 0; mi < 4; ++mi)
#pragma unroll
        for (int ni = 0; ni < 2; ++ni)
            acc[mi][ni] = (v8f)0.0f;

#if defined(USE_TDM)
    // Issue chunk 0 into buffer 0 (one wave; TDM ignores EXEC, tracked by
    // TENSORcnt on the issuing wave only).
    if (tid < 32) {
        const unsigned a_off = (unsigned)(uintptr_t)(void*)&lds[0];
        const unsigned b_off = (unsigned)(uintptr_t)(void*)&lds[TILE_F];
        tdm_load_tile(a_off, x + (size_t)m0 * IN_DIM, B_ROWS);
        tdm_load_tile(b_off, W + (size_t)n0 * IN_DIM, HID_DIM);
    }
#else
    const int lr = tid >> 1;             // manual loader: 2 threads/row
    const int lc = (tid & 1) * 16;
#endif

    int buf = 0;
    for (int kc = 0; kc < IN_DIM; kc += KC) {
        float* As = lds + buf * 2 * TILE_F;
        float* Bs = As + TILE_F;

#if defined(USE_TDM)
        if (tid < 32) __builtin_amdgcn_s_wait_tensorcnt(0);  // chunk ready
        __syncthreads();
        // Prefetch next chunk into the other buffer while we compute.
        if (tid < 32 && kc + KC < IN_DIM) {
            float* An = lds + (buf ^ 1) * 2 * TILE_F;
            const unsigned a_off = (unsigned)(uintptr_t)(void*)An;
            const unsigned b_off = (unsigned)(uintptr_t)(void*)(An + TILE_F);
            tdm_load_tile(a_off, x + (size_t)m0 * IN_DIM + (kc + KC), B_ROWS);
            tdm_load_tile(b_off, W + (size_t)n0 * IN_DIM + (kc + KC), HID_DIM);
        }
#else
        {
            const float4* xg = (const float4*)(x + (size_t)(m0 + lr) * IN_DIM + kc + lc);
            float* as = &As[lr * LDSS + lc];
#pragma unroll
            for (int v = 0; v < 4; ++v) {
                float4 t = xg[v];
                as[4 * v + 0] = t.x; as[4 * v + 1] = t.y;
                as[4 * v + 2] = t.z; as[4 * v + 3] = t.w;
            }
            const float4* wg = (const float4*)(W + (size_t)(n0 + lr) * IN_DIM + kc + lc);
            float* bs = &Bs[lr * LDSS + lc];
#pragma unroll
            for (int v = 0; v < 4; ++v) {
                float4 t = wg[v];
                bs[4 * v + 0] = t.x; bs[4 * v + 1] = t.y;
                bs[4 * v + 2] = t.z; bs[4 * v + 3] = t.w;
            }
            __syncthreads();
        }
#endif

#pragma unroll
        for (int kk = 0; kk < KC; kk += 4) {
            v2f a[4], b[2];
#pragma unroll
            for (int mi = 0; mi < 4; ++mi)
                a[mi] = *(const v2f*)&As[(wm * 64 + mi * 16 + lrow) * LDSS + kk + koff];
#pragma unroll
            for (int ni = 0; ni < 2; ++ni)
                b[ni] = *(const v2f*)&Bs[(wn * 32 + ni * 16 + lrow) * LDSS + kk + koff];
#pragma unroll
            for (int mi = 0; mi < 4; ++mi)
#pragma unroll
                for (int ni = 0; ni < 2; ++ni)
                    acc[mi][ni] = __builtin_amdgcn_wmma_f32_16x16x4_f32(
                        false, a[mi], false, b[ni], (short)0, acc[mi][ni],
                        false, false);
        }
        __syncthreads();   // everyone done reading this buffer
        buf ^= 1;
    }

    // Epilogue: + hb[n].  C layout: VGPR r -> M=r (lane<16) / M=r+8 (lane>=16),
    // N = lane%16.
    const int mhalf = (lane < 16) ? 0 : 8;
#pragma unroll
    for (int ni = 0; ni < 2; ++ni) {
        const int ncol = n0 + wn * 32 + ni * 16 + lrow;
        const float lb = hb[ncol];
#pragma unroll
        for (int mi = 0; mi < 4; ++mi) {
            const int mbase = m0 + wm * 64 + mi * 16 + mhalf;
#pragma unroll
            for (int r = 0; r < 8; ++r) {
                pre_act[(size_t)(mbase + r) * HID_DIM + ncol] = acc[mi][ni][r] + lb;
            }
        }
    }
}

// ------------------- Kernel 2: top-32 + scatter + sparse decode ------------

__global__ __launch_bounds__(256)
void sae_topk_decode(const float* __restrict__ pre_act,
                     const float* __restrict__ W,
                     const float* __restrict__ pre_bias,
                     float* __restrict__ latents,
                     float* __restrict__ recon)
{
    extern __shared__ float smem[];
    float* row     = smem;                           // 16384
    float* rval    = smem + HID_DIM;                 // 256
    int*   ridx    = (int*)(smem + HID_DIM + 256);   // 256
    float* sel_val = smem + HID_DIM + 512;           // 32
    int*   sel_idx = (int*)(smem + HID_DIM + 544);   // 32

    const int tid = threadIdx.x;
    const int b   = blockIdx.x;

    const float* pr  = pre_act + (size_t)b * HID_DIM;
    float*       lat = latents + (size_t)b * HID_DIM;

    {
        const float4* pr4  = (const float4*)pr;
        float4*       row4 = (float4*)row;
        float4*       lat4 = (float4*)lat;
        const float4  z4   = make_float4(0.f, 0.f, 0.f, 0.f);
        for (int i = tid; i < HID_DIM / 4; i += 256) {
            row4[i] = pr4[i];
            lat4[i] = z4;
        }
    }
    __syncthreads();

    for (int it = 0; it < TOPK; ++it) {
        float best = -INFINITY;
        int   bidx = 0;
#pragma unroll 8
        for (int j = 0; j < HID_DIM / 256; ++j) {
            const int idx = tid + j * 256;
            const float v = row[idx];
            if (v > best) { best = v; bidx = idx; }
        }
        rval[tid] = best;
        ridx[tid] = bidx;
        __syncthreads();
        for (int s = 128; s > 0; s >>= 1) {
            if (tid < s) {
                const float v2 = rval[tid + s];
                const int   i2 = ridx[tid + s];
                if (v2 > rval[tid] || (v2 == rval[tid] && i2 < ridx[tid])) {
                    rval[tid] = v2;
                    ridx[tid] = i2;
                }
            }
            __syncthreads();
        }
        if (tid == 0) {
            sel_val[it] = rval[0];
            sel_idx[it] = ridx[0];
            row[ridx[0]] = -INFINITY;
        }
        __syncthreads();
    }

    if (tid < TOPK) {
        const float v = sel_val[tid];
        lat[sel_idx[tid]] = v > 0.f ? v : 0.f;
    }
    __syncthreads();

    for (int d = tid; d < IN_DIM; d += 256) {
        float acc = pre_bias[d];
#pragma unroll
        for (int j = 0; j < TOPK; ++j) {
            float v = sel_val[j];
            v = v > 0.f ? v : 0.f;
            acc += v * W[(size_t)sel_idx[j] * IN_DIM + d];
        }
        recon[(size_t)b * IN_DIM + d] = acc;
    }
}

// ------------------------------- launcher ----------------------------------

extern "C" void kernel_launch(void* const* d_in, const int* in_sizes, int n_in,
                              void* d_out, int out_size, void* d_ws, size_t ws_size,
                              hipStream_t stream) {
    (void)in_sizes; (void)n_in; (void)out_size; (void)ws_size;

    const float* x           = (const float*)d_in[0];   // [8192, 768]
    const float* W_enc       = (const float*)d_in[1];   // [16384, 768]
    const float* pre_bias    = (const float*)d_in[2];   // [768]
    const float* latent_bias = (const float*)d_in[3];   // [16384]

    float* pre_act = (float*)d_out;                               // [8192,16384]
    float* latents = pre_act + (size_t)B_ROWS * HID_DIM;          // [8192,16384]
    float* recon   = latents + (size_t)B_ROWS * HID_DIM;          // [8192,768]
    float* hb      = (float*)d_ws;                                // [16384]

    // 0) fold pre_bias into a per-latent constant
    sae_hbias<<<dim3(HID_DIM / 8), 256, 0, stream>>>(W_enc, pre_bias,
                                                     latent_bias, hb);

    // 1) encode GEMM (WMMA f32, TDM double-buffered LDS staging)
    dim3 grid1(B_ROWS / BM, HID_DIM / BN, 1);
    const size_t lds1 = (size_t)(4 * TILE_F) * sizeof(float);     // 2 bufs x (A+B)
    sae_encode_wmma<<<grid1, 256, lds1, stream>>>(x, W_enc, hb, pre_act);

    // 2) top-k + scatter + sparse decode
    const size_t lds2 = (size_t)(HID_DIM + 512 + 64) * sizeof(float);
    sae_topk_decode<<<dim3(B_ROWS), 256, lds2, stream>>>(
        pre_act, W_enc, pre_bias, latents, recon);
}